// MinibatchLayer_85246510891072
// MI455X (gfx1250) — compile-verified
//
#include <hip/hip_runtime.h>
#include <hip/hip_bf16.h>

// ---------------------------------------------------------------------------
// MinibatchLayer for MI455X (gfx1250, wave32).
//   Stage 1: M[1024][256] = x[1024][1024] @ T[1024][256]   (f32 WMMA 16x16x4)
//   Stage 2: o_b[j][o] = sum_i exp(-sum_k |M[i][o*8+k]-M[j][o*8+k]|) - 1
//   Output:  out[j][0:1024] = x[j][:];  out[j][1024+o] = o_b[j][o]
// ---------------------------------------------------------------------------

typedef __attribute__((ext_vector_type(2))) float v2f;
typedef __attribute__((ext_vector_type(8))) float v8f;

#define N_ROWS   1024   // batch N
#define IN_F     1024
#define OUT_COLS 256    // OUT_F * KD = 32*8
#define OUT_STRIDE 1056 // IN_F + OUT_F

// One wave (32 lanes) computes one 16x16 tile of M = x @ T.
// grid = (OUT_COLS/16, N_ROWS/16) = (16, 64), block = 32 (one wave).
//
// V_WMMA_F32_16X16X4_F32: A is 16x4 (2 VGPRs), B is 4x16 (2 VGPRs), C/D 16x16 f32.
// A layout (ISA 7.12.2): lanes 0-15 hold M=0..15; v0 = K=(0|2), v1 = K=(1|3)
//   with the +2 K-offset selected by lane>=16.  B mirrors it across lanes.
// D layout: VGPR r, lanes 0-15 -> row M=r, lanes 16-31 -> row M=r+8, col=lane&15.
__global__ __launch_bounds__(32)
void mb_gemm_wmma_f32(const float* __restrict__ x,
                      const float* __restrict__ T,
                      float* __restrict__ M)
{
    const int lane = threadIdx.x;        // 0..31
    const int half = lane >> 4;          // 0 or 1
    const int l    = lane & 15;          // 0..15
    const int n0   = blockIdx.x * 16;    // output column tile
    const int m0   = blockIdx.y * 16;    // output row tile

    // A: row (m0+l) of x, starting at K offset 2*half (float2 per step)
    const float* arow = x + (size_t)(m0 + l) * IN_F + 2 * half;
    // B: column (n0+l) of T, rows (kk + 2*half) and (kk + 2*half + 1)
    const float* bcol = T + (size_t)(2 * half) * OUT_COLS + (n0 + l);

    v8f acc = {};

#pragma unroll 8
    for (int kk = 0; kk < IN_F; kk += 4) {
        v2f a;
        {
            // 8-byte aligned contiguous pair -> global_load_b64
            float2 av = *(const float2*)(arow + kk);
            a.x = av.x;
            a.y = av.y;
        }
        v2f b;
        b.x = bcol[(size_t)kk * OUT_COLS];
        b.y = bcol[(size_t)(kk + 1) * OUT_COLS];

        // (neg_a, A, neg_b, B, c_mod, C, reuse_a, reuse_b)
        acc = __builtin_amdgcn_wmma_f32_16x16x4_f32(
            false, a, false, b, (short)0, acc, false, false);
    }

    // Store D: VGPR r holds row (m0 + r + 8*half), column (n0 + l).
    float* drow = M + (size_t)m0 * OUT_COLS + n0 + l;
#pragma unroll
    for (int r = 0; r < 8; ++r) {
        drow[(size_t)(r + 8 * half) * OUT_COLS] = acc[r];
    }
}

// One block per output row j. 256 threads = (o = t&31) x (group g = t>>5).
// Each thread sums exp(-L1) over 128 i values; LDS reduces the 8 groups.
// Also copies the xf row into out[j][0:1024] (one float4 per thread).
__global__ __launch_bounds__(256)
void mb_pairwise_exp(const float* __restrict__ x,
                     const float* __restrict__ M,
                     float* __restrict__ out)
{
    const int j = blockIdx.x;
    const int t = threadIdx.x;

    // ---- pass-through copy: out[j][0:1024] = x[j][:] ----
    // j*OUT_STRIDE floats = j*4224 bytes, 16B aligned -> float4 ok.
    {
        const float4* xin = (const float4*)(x + (size_t)j * IN_F);
        float4*       xo  = (float4*)(out + (size_t)j * OUT_STRIDE);
        xo[t] = xin[t];   // 256 threads x 16B = 4KB row
    }

    const int o = t & 31;   // output feature
    const int g = t >> 5;   // i-range group, 0..7

    // M[j] fragment for this o (8 floats, 32B aligned)
    float mj[8];
    {
        const float4* Mj = (const float4*)(M + (size_t)j * OUT_COLS + o * 8);
        float4 m0 = Mj[0];
        float4 m1 = Mj[1];
        mj[0] = m0.x; mj[1] = m0.y; mj[2] = m0.z; mj[3] = m0.w;
        mj[4] = m1.x; mj[5] = m1.y; mj[6] = m1.z; mj[7] = m1.w;
    }

    float accv = 0.0f;
    const int i0 = g * 128;
#pragma unroll 4
    for (int i = i0; i < i0 + 128; ++i) {
        const float* Mi = M + (size_t)i * OUT_COLS + o * 8;
        float4 a0 = *(const float4*)(Mi);       // b128 load
        float4 a1 = *(const float4*)(Mi + 4);   // b128 load
        float s = __builtin_fabsf(a0.x - mj[0]) + __builtin_fabsf(a0.y - mj[1])
                + __builtin_fabsf(a0.z - mj[2]) + __builtin_fabsf(a0.w - mj[3])
                + __builtin_fabsf(a1.x - mj[4]) + __builtin_fabsf(a1.y - mj[5])
                + __builtin_fabsf(a1.z - mj[6]) + __builtin_fabsf(a1.w - mj[7]);
        accv += __expf(-s);   // v_exp_f32
    }

    __shared__ float red[256];
    red[t] = accv;
    __syncthreads();

    if (g == 0) {
        float s = 0.0f;
#pragma unroll
        for (int gg = 0; gg < 8; ++gg)
            s += red[gg * 32 + o];
        // self term i==j contributes exp(0)=1; remove it.
        out[(size_t)j * OUT_STRIDE + IN_F + o] = s - 1.0f;
    }
}

extern "C" void kernel_launch(void* const* d_in, const int* in_sizes, int n_in,
                              void* d_out, int out_size, void* d_ws, size_t ws_size,
                              hipStream_t stream)
{
    const float* x = (const float*)d_in[0];   // [1024, 1024] f32
    const float* T = (const float*)d_in[1];   // [1024, 32, 8] f32 == [1024, 256] row-major
    float* out = (float*)d_out;               // [1024, 1056] f32
    float* M   = (float*)d_ws;                // [1024, 256] f32 scratch (1 MB)

    // Stage 1: WMMA f32 GEMM, one wave per 16x16 tile of M.
    dim3 ggrid(OUT_COLS / 16, N_ROWS / 16);   // (16, 64)
    mb_gemm_wmma_f32<<<ggrid, 32, 0, stream>>>(x, T, M);

    // Stage 2: pairwise L1 + exp reduction, one block per row j.
    mb_pairwise_exp<<<N_ROWS, 256, 0, stream>>>(x, M, out);
}